// DGCF_4269197492543
// MI455X (gfx1250) — compile-verified
//
#include <hip/hip_runtime.h>

// ---- problem constants (match reference) ----
#define NUSERS   100000
#define NITEMS   50000
#define NNODES   150000           // NUSERS + NITEMS
#define EMB      64
#define NFAC     4
#define DF       16
#define EDGES    2000000
#define NLAYERS  2
#define NITER    2
#define L2EPS    1e-12f

typedef __attribute__((ext_vector_type(2))) float v2f;
typedef __attribute__((ext_vector_type(8))) float v8f;

// tanh(x) = 1 - 2/(exp(2x)+1) using native V_EXP_F32 + V_RCP_F32 (TRANS ops,
// co-execute with the WMMA pipe).  Exact at the saturation limits; input here
// is an l2-normalized component so |2x*log2e| <= 2.89 (well-conditioned).
__device__ __forceinline__ float fast_tanhf(float x) {
  float e = __builtin_amdgcn_exp2f(x * 2.885390081777927f);   // exp(2x)
  return 1.0f - 2.0f * __builtin_amdgcn_rcpf(e + 1.0f);
}

static inline int cdiv(long long a, long long b) { return (int)((a + b - 1) / b); }

// ---------------------------------------------------------------------------
// init: ego = concat(user, item); accum = ego (all_embeds[0])
__global__ __launch_bounds__(256) void k_init_ego(
    const float* __restrict__ ue, const float* __restrict__ ie,
    float* __restrict__ ego, float* __restrict__ acc) {
  int i = blockIdx.x * blockDim.x + threadIdx.x;
  if (i >= NNODES * EMB) return;
  float v = (i < NUSERS * EMB) ? ue[i] : ie[i - NUSERS * EMB];
  ego[i] = v;
  acc[i] = v;
}

__global__ __launch_bounds__(256) void k_init_A(float* __restrict__ A) {
  int i = blockIdx.x * blockDim.x + threadIdx.x;
  if (i < EDGES * NFAC) A[i] = 1.0f;
}

// ---------------------------------------------------------------------------
// softmax over the 4 factors of each edge.  A,S layout: [E][4] (float4/edge)
__global__ __launch_bounds__(256) void k_edge_softmax(
    const float* __restrict__ A, float* __restrict__ S) {
  int e = blockIdx.x * blockDim.x + threadIdx.x;
  if (e >= EDGES) return;
  float4 a = ((const float4*)A)[e];
  float m  = fmaxf(fmaxf(a.x, a.y), fmaxf(a.z, a.w));
  const float L2E = 1.4426950408889634f;
  float e0 = __builtin_amdgcn_exp2f((a.x - m) * L2E);
  float e1 = __builtin_amdgcn_exp2f((a.y - m) * L2E);
  float e2 = __builtin_amdgcn_exp2f((a.z - m) * L2E);
  float e3 = __builtin_amdgcn_exp2f((a.w - m) * L2E);
  float inv = __builtin_amdgcn_rcpf(e0 + e1 + e2 + e3);
  ((float4*)S)[e] = make_float4(e0 * inv, e1 * inv, e2 * inv, e3 * inv);
}

// ---------------------------------------------------------------------------
// rowsum[n][f] += scores[e][f] for h[e]==n   (rowsum layout [N][4])
__global__ __launch_bounds__(256) void k_rowsum(
    const float* __restrict__ S, const int* __restrict__ h,
    float* __restrict__ rs) {
  int tid = blockIdx.x * blockDim.x + threadIdx.x;
  if (tid >= EDGES * NFAC) return;
  int f = tid & 3;
  int e = tid >> 2;
  unsafeAtomicAdd(rs + (size_t)h[e] * NFAC + f, S[tid]);
}

// d_col = 1/sqrt(rowsum)
__global__ __launch_bounds__(256) void k_dcol(
    const float* __restrict__ rs, float* __restrict__ dc) {
  int i = blockIdx.x * blockDim.x + threadIdx.x;
  if (i >= NNODES * NFAC) return;
  dc[i] = 1.0f / sqrtf(rs[i]);
}

// ---------------------------------------------------------------------------
// out[h] += score * dcol[t] * x[t]   (16 lanes per (edge,factor), coalesced 64B)
__global__ __launch_bounds__(256) void k_propagate(
    const float* __restrict__ x, const float* __restrict__ S,
    const float* __restrict__ dc, const int* __restrict__ h,
    const int* __restrict__ t, float* __restrict__ out) {
  int tid = blockIdx.x * blockDim.x + threadIdx.x;     // EDGES*NFAC*DF = 128M
  int j   = tid & 15;                                  // df lane
  int idx = tid >> 4;                                  // e*4+f
  int f   = idx & 3;
  int e   = idx >> 2;
  int tt = t[e], hh = h[e];
  float val = S[idx] * dc[(size_t)tt * NFAC + f] * x[(size_t)tt * EMB + f * DF + j];
  unsafeAtomicAdd(out + (size_t)hh * EMB + f * DF + j, val);
  // CDNA5 global_prefetch_b8 to warm a future gather row through GL2
  if ((tid & 255) == 0) {
    int ep = e + 1024;
    if (ep < EDGES) __builtin_prefetch(x + (size_t)t[ep] * EMB, 0, 1);
  }
}

// out[n] *= dcol[n]  (left D_col application)
__global__ __launch_bounds__(256) void k_scale(
    float* __restrict__ out, const float* __restrict__ dc) {
  int i = blockIdx.x * blockDim.x + threadIdx.x;
  if (i >= NNODES * EMB) return;
  int c = i & 63, n = i >> 6;
  out[i] *= dc[(size_t)n * NFAC + (c >> 4)];
}

// ---------------------------------------------------------------------------
// routing update via WMMA:  A[e][f] += diag( H · T^T )
//   H = l2norm(out[h[e]] factor f)   (16 edges x 16 df)
//   T = tanh(l2norm(x[t[e]] factor f))
// One wave32 handles one (factor, 16-edge block).  V_WMMA_F32_16X16X4_F32 x4
// accumulates over K=16.  A/B operands live entirely in-lane (row = lane%16).
__global__ __launch_bounds__(256) void k_routing_wmma(
    const float* __restrict__ out, const float* __restrict__ x,
    const int* __restrict__ h, const int* __restrict__ t,
    float* __restrict__ A) {
  int tid  = blockIdx.x * blockDim.x + threadIdx.x;
  int lane = threadIdx.x & 31;
  int w    = tid >> 5;            // global wave id, exactly NFAC*EDGES/16 waves
  int f    = w & 3;
  int eb   = w >> 2;              // 16-edge block id
  int r    = lane & 15;
  int e    = eb * 16 + r;

  int hh = h[e], tt = t[e];
  const float4* orow = (const float4*)(out + (size_t)hh * EMB + f * DF);
  const float4* xrow = (const float4*)(x   + (size_t)tt * EMB + f * DF);

  float H[16], T[16];
  {
    float4 v0 = orow[0], v1 = orow[1], v2 = orow[2], v3 = orow[3];
    H[0]=v0.x; H[1]=v0.y; H[2]=v0.z; H[3]=v0.w;
    H[4]=v1.x; H[5]=v1.y; H[6]=v1.z; H[7]=v1.w;
    H[8]=v2.x; H[9]=v2.y; H[10]=v2.z; H[11]=v2.w;
    H[12]=v3.x; H[13]=v3.y; H[14]=v3.z; H[15]=v3.w;
    float s = 0.f;
#pragma unroll
    for (int j = 0; j < 16; ++j) s += H[j] * H[j];
    float invn = __builtin_amdgcn_rcpf(fmaxf(sqrtf(s), L2EPS));
#pragma unroll
    for (int j = 0; j < 16; ++j) H[j] *= invn;
  }
  {
    float4 v0 = xrow[0], v1 = xrow[1], v2 = xrow[2], v3 = xrow[3];
    T[0]=v0.x; T[1]=v0.y; T[2]=v0.z; T[3]=v0.w;
    T[4]=v1.x; T[5]=v1.y; T[6]=v1.z; T[7]=v1.w;
    T[8]=v2.x; T[9]=v2.y; T[10]=v2.z; T[11]=v2.w;
    T[12]=v3.x; T[13]=v3.y; T[14]=v3.z; T[15]=v3.w;
    float s = 0.f;
#pragma unroll
    for (int j = 0; j < 16; ++j) s += T[j] * T[j];
    float invn = __builtin_amdgcn_rcpf(fmaxf(sqrtf(s), L2EPS));
#pragma unroll
    for (int j = 0; j < 16; ++j) T[j] = fast_tanhf(T[j] * invn);
  }

  // A-matrix 16x4 f32: lanes<16 carry K=k,k+1; lanes>=16 carry K=k+2,k+3.
  // B-matrix 4x16 has the same K split; row index = lane%16 for both since
  // B[kk][n] = T[n][k+kk].  Branchless select keeps EXEC all-ones for WMMA.
  bool hi = lane >= 16;
  v8f c = {};
#pragma unroll
  for (int k = 0; k < 16; k += 4) {
    v2f a, b;
    a.x = hi ? H[k + 2] : H[k + 0];
    a.y = hi ? H[k + 3] : H[k + 1];
    b.x = hi ? T[k + 2] : T[k + 0];
    b.y = hi ? T[k + 3] : T[k + 1];
    c = __builtin_amdgcn_wmma_f32_16x16x4_f32(
        /*neg_a=*/false, a, /*neg_b=*/false, b,
        /*c_mod=*/(short)0, c, /*reuse_a=*/false, /*reuse_b=*/false);
  }

  // diag(C): (i,i) lives in VGPR (i<8 ? i : i-8) at lane (i<8 ? i : 16+i)
  float myDot = 0.f;
#pragma unroll
  for (int i = 0; i < 16; ++i) {
    float di = __shfl(c[(i < 8) ? i : (i - 8)], (i < 8) ? i : (16 + i), 32);
    if (r == i) myDot = di;
  }
  if (lane < 16) A[(size_t)e * NFAC + f] += myDot;
}

// ---------------------------------------------------------------------------
__global__ __launch_bounds__(256) void k_accum(
    float* __restrict__ acc, const float* __restrict__ src) {
  int i = blockIdx.x * blockDim.x + threadIdx.x;
  if (i < NNODES * EMB) acc[i] += src[i];
}

__global__ __launch_bounds__(256) void k_final(
    const float* __restrict__ acc, float* __restrict__ outp) {
  int i = blockIdx.x * blockDim.x + threadIdx.x;
  if (i < NNODES * EMB) outp[i] = acc[i] * (1.0f / (1 + NLAYERS));
}

// ---------------------------------------------------------------------------
extern "C" void kernel_launch(void* const* d_in, const int* in_sizes, int n_in,
                              void* d_out, int out_size, void* d_ws, size_t ws_size,
                              hipStream_t stream) {
  const float* user_emb = (const float*)d_in[0];
  const float* item_emb = (const float*)d_in[1];
  const int*   h_list   = (const int*)d_in[2];
  const int*   t_list   = (const int*)d_in[3];

  const long long NE = (long long)NNODES * EMB;   // 9.6M floats
  const long long E4 = (long long)EDGES * NFAC;   // 8M floats
  const long long N4 = (long long)NNODES * NFAC;  // 0.6M floats

  float* B0  = (float*)d_ws;          // ego / ping
  float* B1  = B0  + NE;              // out / pong
  float* acc = B1  + NE;              // running sum of all_embeds
  float* Av  = acc + NE;              // A_values  [E][4]
  float* Sc  = Av  + E4;              // scores    [E][4]
  float* rs  = Sc  + E4;              // rowsum    [N][4]
  float* dc  = rs  + N4;              // d_col     [N][4]

  k_init_ego<<<cdiv(NE, 256), 256, 0, stream>>>(user_emb, item_emb, B0, acc);
  k_init_A  <<<cdiv(E4, 256), 256, 0, stream>>>(Av);

  float* xb = B0;   // layer input
  float* ob = B1;   // layer output
  for (int layer = 0; layer < NLAYERS; ++layer) {
    for (int it = 0; it < NITER; ++it) {
      k_edge_softmax<<<cdiv(EDGES, 256), 256, 0, stream>>>(Av, Sc);
      hipMemsetAsync(rs, 0, N4 * sizeof(float), stream);
      k_rowsum<<<cdiv(E4, 256), 256, 0, stream>>>(Sc, h_list, rs);
      k_dcol  <<<cdiv(N4, 256), 256, 0, stream>>>(rs, dc);
      hipMemsetAsync(ob, 0, NE * sizeof(float), stream);
      k_propagate<<<cdiv((long long)EDGES * EMB, 256), 256, 0, stream>>>(
          xb, Sc, dc, h_list, t_list, ob);
      k_scale<<<cdiv(NE, 256), 256, 0, stream>>>(ob, dc);
      // waves = NFAC * EDGES/16 = 500000 (exact) -> 62500 blocks of 8 waves
      k_routing_wmma<<<(NFAC * (EDGES / 16) * 32) / 256, 256, 0, stream>>>(
          ob, xb, h_list, t_list, Av);
    }
    k_accum<<<cdiv(NE, 256), 256, 0, stream>>>(acc, ob);
    float* tmp = xb; xb = ob; ob = tmp;   // next layer's input is this output
  }
  k_final<<<cdiv(NE, 256), 256, 0, stream>>>(acc, (float*)d_out);
}